// TorchModelMultiGeo_40836549050714
// MI455X (gfx1250) — compile-verified
//
#include <hip/hip_runtime.h>
#include <math.h>

typedef __attribute__((ext_vector_type(2))) float v2f;
typedef __attribute__((ext_vector_type(8))) float v8f;

#define F  16
#define NV 4096
#define NT 512
#define MPTS 4096
#define SCALE 0.1f

#define MT (MPTS/16)            // 256 model tiles of 16 points
#define BBUF_FLOATS (MT*64)     // 16384 floats = 64 KB

// workspace layout (in floats)
#define WS_TS   0               // 16*3 translations
#define WS_RS   64              // 16*9 rotations (row-major)
#define WS_BBUF 256             // model B-layout buffer

// ---------------- Rodrigues ----------------
__device__ inline void rot_from_omega(const float* w, float dt, float* R) {
    float theta = sqrtf(w[0]*w[0] + w[1]*w[1] + w[2]*w[2]);
    float inv   = 1.0f / fmaxf(theta, 1e-8f);
    float ax = w[0]*inv, ay = w[1]*inv, az = w[2]*inv;
    float s = sinf(theta*dt), c = 1.0f - cosf(theta*dt);
    float S[9]  = {0.f,-az,ay,  az,0.f,-ax,  -ay,ax,0.f};
    float S2[9];
    for (int i=0;i<3;i++) for (int j=0;j<3;j++){
        float acc=0.f; for(int k=0;k<3;k++) acc += S[i*3+k]*S[k*3+j];
        S2[i*3+j]=acc;
    }
    for (int i=0;i<9;i++) R[i] = s*S[i] + c*S2[i];
    R[0]+=1.f; R[4]+=1.f; R[8]+=1.f;
}

// ---------------- sequential pose scan (1 thread) ----------------
__global__ void pose_kernel(const float* __restrict__ state,
                            const float* __restrict__ phys,
                            const float* __restrict__ dts,
                            float* __restrict__ ws, float* __restrict__ out) {
    if (threadIdx.x != 0 || blockIdx.x != 0) return;
    out[0] = 0.0f;
    float t[3] = {state[0], state[1], state[2]};
    float w0[3] = {state[3], state[4], state[5]};
    float r[9];
    rot_from_omega(w0, 1.0f, r);
    float* ts = ws + WS_TS;
    float* rs = ws + WS_RS;
    for (int i=0;i<3;i++) ts[i]=t[i];
    for (int i=0;i<9;i++) rs[i]=r[i];
    for (int fidx=1; fidx<F; ++fidx) {
        float dt = dts[fidx];
        const float* row = phys + fidx*12;
        for (int i=0;i<3;i++) t[i] += row[6+i]*dt;
        float om[3] = {row[9], row[10], row[11]};
        float Rs[9], rn[9];
        rot_from_omega(om, dt, Rs);
        for (int i=0;i<3;i++) for (int j=0;j<3;j++){
            float acc=0.f; for(int k=0;k<3;k++) acc += Rs[i*3+k]*r[k*3+j];
            rn[i*3+j]=acc;
        }
        for (int i=0;i<9;i++) r[i]=rn[i];
        for (int i=0;i<3;i++) ts[fidx*3+i]=t[i];
        for (int i=0;i<9;i++) rs[fidx*9+i]=r[i];
    }
}

// ---------------- model B-matrix layout prep ----------------
// Per 16-point tile (64 floats): [n*2]=x, [n*2+1]=y, [32+n*2]=z, [32+n*2+1]=|m|^2
// so lane L (L<16) loads (x,y) for col n=L, lane L+16 loads (z,|m|^2).
__global__ void prep_model(const float* __restrict__ mp, float* __restrict__ bbuf) {
    int j = blockIdx.x*blockDim.x + threadIdx.x;
    if (j >= MPTS) return;
    float x = mp[j*3+0], y = mp[j*3+1], z = mp[j*3+2];
    float msq = x*x + y*y + z*z;
    int tile = j >> 4, n = j & 15;
    float* t = bbuf + tile*64;
    t[n*2+0]    = x;
    t[n*2+1]    = y;
    t[32+n*2+0] = z;
    t[32+n*2+1] = msq;
}

// ---------------- WMMA chamfer ----------------
__global__ __launch_bounds__(256) void chamfer_kernel(
    const float* __restrict__ vp, const float* __restrict__ tp,
    const float* __restrict__ ws, float* __restrict__ out)
{
    __shared__ float sB[BBUF_FLOATS];   // 64 KB model tiles (of 320 KB WGP LDS)
    const float* bbuf = ws + WS_BBUF;
    for (int i = threadIdx.x; i < BBUF_FLOATS/4; i += blockDim.x)
        ((float4*)sB)[i] = ((const float4*)bbuf)[i];
    __syncthreads();

    const int wave = (int)((blockIdx.x * blockDim.x + threadIdx.x) >> 5);
    const int lane = (int)(threadIdx.x & 31);

    // row-tile mapping: 16 frames * 256 visual tiles, then 16 * 32 tactile tiles
    const int VIS_TILES = F * (NV/16);     // 4096
    const float* src; int f, baserow; float weight;
    if (wave < VIS_TILES) {
        f       = wave >> 8;
        baserow = (wave & 255) << 4;
        src     = vp + f*NV*3;
        weight  = 1.0f / (float)NV;
    } else {
        int w2  = wave - VIS_TILES;
        f       = w2 >> 5;
        baserow = (w2 & 31) << 4;
        src     = tp + f*NT*3;
        weight  = 0.1f / (float)NT;
    }

    const float* t = ws + WS_TS + f*3;
    const float* r = ws + WS_RS + f*9;
    const int row = baserow + (lane & 15);

    float px = src[row*3+0] - t[0];
    float py = src[row*3+1] - t[1];
    float pz = src[row*3+2] - t[2];
    const float inv = 1.0f / SCALE;
    float pc0 = (px*r[0] + py*r[3] + pz*r[6]) * inv;
    float pc1 = (px*r[1] + py*r[4] + pz*r[7]) * inv;
    float pc2 = (px*r[2] + py*r[5] + pz*r[8]) * inv;
    float pcsq = pc0*pc0 + pc1*pc1 + pc2*pc2;

    // A(16x4): row i = (-2x, -2y, -2z, 1)  -> D = -2 p.m + |m|^2
    v2f a;
    if (lane < 16) { a.x = -2.0f*pc0; a.y = -2.0f*pc1; }
    else           { a.x = -2.0f*pc2; a.y = 1.0f;      }

    v8f runmin;
#pragma unroll
    for (int k=0;k<8;k++) runmin[k] = 3.0e38f;
    v8f zero = {};

    const int boff = ((lane < 16) ? 0 : 32) + (lane & 15)*2;
#pragma unroll 4
    for (int jt = 0; jt < MT; ++jt) {
        v2f b = *(const v2f*)(sB + jt*64 + boff);     // 8B aligned LDS load
        v8f d = __builtin_amdgcn_wmma_f32_16x16x4_f32(
            /*neg_a=*/false, a, /*neg_b=*/false, b,
            /*c_mod=*/(short)0, zero, /*reuse_a=*/false, /*reuse_b=*/false);
#pragma unroll
        for (int k=0;k<8;k++) runmin[k] = fminf(runmin[k], d[k]);
    }

    // min over the 16 column lanes of each wave32 half (masks 1,2,4,8 stay in-half)
#pragma unroll
    for (int mask = 1; mask <= 8; mask <<= 1) {
#pragma unroll
        for (int k=0;k<8;k++) {
            float o = __shfl_xor(runmin[k], mask, 32);
            runmin[k] = fminf(runmin[k], o);
        }
    }

    // add |p_i|^2 per row: VGPR k is row k (lanes<16) or k+8 (lanes>=16);
    // pcsq for row r lives in lane r
    float acc = 0.0f;
    const int half = (lane & 16) ? 8 : 0;
#pragma unroll
    for (int k=0;k<8;k++) {
        float ps = __shfl(pcsq, k + half, 32);
        acc += runmin[k] + ps;
    }
    if ((lane & 15) == 0)
        atomicAdd(out, weight * acc);
}

// ---------------- launch ----------------
extern "C" void kernel_launch(void* const* d_in, const int* in_sizes, int n_in,
                              void* d_out, int out_size, void* d_ws, size_t ws_size,
                              hipStream_t stream) {
    const float* state = (const float*)d_in[0];
    const float* model = (const float*)d_in[1];
    const float* vp    = (const float*)d_in[2];
    const float* tp    = (const float*)d_in[3];
    const float* phys  = (const float*)d_in[4];
    const float* dts   = (const float*)d_in[5];
    float* out = (float*)d_out;
    float* ws  = (float*)d_ws;

    pose_kernel<<<1, 32, 0, stream>>>(state, phys, dts, ws, out);
    prep_model<<<MPTS/256, 256, 0, stream>>>(model, ws + WS_BBUF);

    const int TOTAL_WAVES = F*(NV/16) + F*(NT/16);   // 4608
    const int BLOCKS = (TOTAL_WAVES * 32) / 256;     // 576
    chamfer_kernel<<<BLOCKS, 256, 0, stream>>>(vp, tp, ws, out);
}